// MemoryGraph_5720896438322
// MI455X (gfx1250) — compile-verified
//
#include <hip/hip_runtime.h>
#include <math.h>

typedef __attribute__((ext_vector_type(2))) float v2f;
typedef __attribute__((ext_vector_type(8))) float v8f;
typedef float v2fu __attribute__((ext_vector_type(2), aligned(4)));   // unaligned-safe 8B load

#define BSZB 8      // batch
#define NN   2048   // nodes
#define KCON 32     // connections per node
#define DD   64     // feature dim
#define HH   128    // hidden (mod/state/msg)
#define MOD_IN   161
#define MOD_OUT  97
#define STATE_IN 129
#define MSG_IN   128

__device__ __forceinline__ float sigmoidf_(float x) {
    return 1.0f / (1.0f + __expf(-x));
}

__device__ __forceinline__ v8f wmma4(v2f a, v2f b, v8f c) {
    return __builtin_amdgcn_wmma_f32_16x16x4_f32(false, a, false, b, (short)0, c, false, false);
}

// out(16-row tile x 8 cols) = W(rows x I, row-major) @ Xs(I x 8 in LDS, [i][16])
// A-frag per ISA 16x4 f32 layout: lanes 0-15 -> K=k+0/k+1, lanes 16-31 -> K=k+2/k+3
// Guard-free body (unaligned b64 A loads) + single guarded tail; dual acc chains.
template<int I>
__device__ __forceinline__ v8f gemm_a_rowmajor(const float* __restrict__ W,
                                               const float* __restrict__ Xs,
                                               int mbase, int lane) {
    constexpr int KBODY = I & ~3;              // k+3 < I guaranteed inside body
    static_assert(KBODY % 8 == 0, "body must unroll by 8");
    const int m    = lane & 15;
    const int koff = (lane >> 4) << 1;
    const int col  = lane & 15;
    const float* Wrow = W + (size_t)(mbase + m) * I;
    v8f acc0 = {}, acc1 = {};
    for (int k = 0; k < KBODY; k += 8) {
        int k0 = k + koff;
        v2f a0 = *(const v2fu*)(Wrow + k0);
        v2f b0; b0.x = Xs[k0 * 16 + col]; b0.y = Xs[(k0 + 1) * 16 + col];
        acc0 = wmma4(a0, b0, acc0);
        int k1 = k0 + 4;
        v2f a1 = *(const v2fu*)(Wrow + k1);
        v2f b1; b1.x = Xs[k1 * 16 + col]; b1.y = Xs[(k1 + 1) * 16 + col];
        acc1 = wmma4(a1, b1, acc1);
    }
    if constexpr (KBODY < I) {                 // tail: at most 3 valid K elements
        int k0 = KBODY + koff;
        v2f a, b;
        a.x = (k0 < I)     ? Wrow[k0]     : 0.0f;
        a.y = (k0 + 1 < I) ? Wrow[k0 + 1] : 0.0f;
        b.x = Xs[k0 * 16 + col];               // Xs zero-padded past I
        b.y = Xs[(k0 + 1) * 16 + col];
        acc0 = wmma4(a, b, acc0);
    }
    return acc0 + acc1;
}

// out(16-row tile x 8 cols) = W^T @ Ys where W is (128 x O) row-major.
// A[m][k] = W[k][m]: 16 contiguous lanes per K-row (coalesced). Row clamp instead of
// guards: invalid output rows compute garbage from a duplicated row, discarded at store.
template<int O>
__device__ __forceinline__ v8f gemm_a_transposed(const float* __restrict__ W,
                                                 const float* __restrict__ Ys,
                                                 int mbase, int lane) {
    const int koff = (lane >> 4) << 1;
    const int col  = lane & 15;
    const int mg   = mbase + (lane & 15);
    const int mc   = (mg < O) ? mg : (O - 1);
    v8f acc0 = {}, acc1 = {};
    for (int k = 0; k < HH; k += 8) {
        int k0 = k + koff;
        v2f a0; a0.x = W[k0 * O + mc]; a0.y = W[(k0 + 1) * O + mc];
        v2f b0; b0.x = Ys[k0 * 16 + col]; b0.y = Ys[(k0 + 1) * 16 + col];
        acc0 = wmma4(a0, b0, acc0);
        int k1 = k0 + 4;
        v2f a1; a1.x = W[k1 * O + mc]; a1.y = W[(k1 + 1) * O + mc];
        v2f b1; b1.x = Ys[k1 * 16 + col]; b1.y = Ys[(k1 + 1) * 16 + col];
        acc1 = wmma4(a1, b1, acc1);
    }
    return acc0 + acc1;
}

__global__ __launch_bounds__(256) void memgraph_kernel(
    const float* __restrict__ h, const float* __restrict__ prev_messages,
    const float* __restrict__ hebb, const float* __restrict__ decay_logit,
    const float* __restrict__ prim,
    const float* __restrict__ mod_w1, const float* __restrict__ mod_b1,
    const float* __restrict__ mod_w2, const float* __restrict__ mod_b2,
    const float* __restrict__ st_w1, const float* __restrict__ st_b1,
    const float* __restrict__ st_w2, const float* __restrict__ st_b2,
    const float* __restrict__ ms_w1, const float* __restrict__ ms_b1,
    const float* __restrict__ ms_w2, const float* __restrict__ ms_b2,
    const float* __restrict__ dbw, const float* __restrict__ dgw,
    const int* __restrict__ conn, float* __restrict__ out)
{
    __shared__ float Xs[176 * 16];   // activations, [i][batch-col], cols 8..15 zero
    __shared__ float Ys[128 * 16];   // hidden (hid / sh / mh), [h][batch-col]
    __shared__ float Ms[112 * 16];   // raw mod_out, [o][batch-col]

    const int n    = blockIdx.x;
    const int tid  = threadIdx.x;
    const int lane = tid & 31;
    const int wid  = tid >> 5;
    const int col  = lane & 15;
    const int hi   = lane >> 4;

    // ---- stage 0: mod_in = [hebbian(32), h(64), decay(1), primitives(64)] -> Xs, zero padded
    for (int idx = tid; idx < 176 * 16; idx += 256) {
        int i = idx >> 4, b = idx & 15;
        float v = 0.0f;
        if (b < BSZB && i < MOD_IN) {
            if (i < KCON)              v = hebb[(size_t)b * NN * KCON + (size_t)n * KCON + i];
            else if (i < KCON + DD)    v = h[(size_t)b * NN * DD + (size_t)n * DD + (i - KCON)];
            else if (i == KCON + DD)   v = decay_logit[(size_t)b * NN + n];
            else                       v = prim[(size_t)b * NN * DD + (size_t)n * DD + (i - KCON - DD - 1)];
        }
        Xs[idx] = v;
    }
    __syncthreads();

    // ---- mod GEMM1: hid = tanh(W1(128x161) @ X + b1)  (8 waves x 16-row tiles)
    {
        const float* W1 = mod_w1 + (size_t)n * HH * MOD_IN;
        const float* B1 = mod_b1 + (size_t)n * HH;
        int mbase = wid * 16;
        v8f acc = gemm_a_rowmajor<MOD_IN>(W1, Xs, mbase, lane);
        for (int r = 0; r < 8; ++r) {
            int row = mbase + r + 8 * hi;
            Ys[row * 16 + col] = tanhf(acc[r] + B1[row]);
        }
    }
    __syncthreads();

    // ---- mod GEMM2: mod_out(97x8) = W2^T @ hid + b2 (raw, activations applied later)
    if (wid < 7) {
        const float* W2 = mod_w2 + (size_t)n * HH * MOD_OUT;
        const float* B2 = mod_b2 + (size_t)n * MOD_OUT;
        int mbase = wid * 16;
        v8f acc = gemm_a_transposed<MOD_OUT>(W2, Ys, mbase, lane);
        for (int r = 0; r < 8; ++r) {
            int row = mbase + r + 8 * hi;
            if (row < MOD_OUT) Ms[row * 16 + col] = acc[r] + B2[row];
        }
    }
    __syncthreads();

    // ---- gather + dendrite tree; rebuild Xs = sx = [received(64), h(64), decay(1)]
    {
        for (int idx = tid; idx < (144 - 129) * 16; idx += 256)   // zero rows 129..143
            Xs[129 * 16 + idx] = 0.0f;
        for (int idx = tid; idx < DD * BSZB; idx += 256) {        // h -> rows 64..127
            int d = idx & 63, b = idx >> 6;
            Xs[(64 + d) * 16 + b] = h[(size_t)b * NN * DD + (size_t)n * DD + d];
        }
        if (tid < BSZB)                                           // decay -> row 128
            Xs[128 * 16 + tid] = sigmoidf_(Ms[32 * 16 + tid]);

        // wave = batch b, lane covers channels d0, d0+1
        int b  = wid;
        int d0 = lane * 2;
        const float* dbwn = dbw + (size_t)n * (4 * 8 * DD);
        const float* dgwn = dgw + (size_t)n * (4 * DD);
        const int*   cn   = conn + (size_t)n * KCON;
        float recv0 = 0.0f, recv1 = 0.0f;
        for (int g = 0; g < 4; ++g) {            // NB=4 branches
            float a0 = 0.0f, a1 = 0.0f;
            for (int s = 0; s < 8; ++s) {        // BSZ=8 per branch
                int kk  = g * 8 + s;
                int idx = cn[kk];
                float ws = sigmoidf_(Ms[kk * 16 + b]);
                const float* mp = prev_messages + ((size_t)b * NN + (size_t)idx) * DD;
                a0 += ws * mp[d0]     * dbwn[kk * DD + d0];
                a1 += ws * mp[d0 + 1] * dbwn[kk * DD + d0 + 1];
            }
            recv0 += tanhf(a0) * dgwn[g * DD + d0];
            recv1 += tanhf(a1) * dgwn[g * DD + d0 + 1];
        }
        Xs[d0 * 16 + b]       = tanhf(recv0);    // NG=1 -> mean == group_out
        Xs[(d0 + 1) * 16 + b] = tanhf(recv1);
    }
    __syncthreads();

    // ---- state GEMM1: sh = tanh(W1(128x129) @ sx + b1)
    {
        const float* W1 = st_w1 + (size_t)n * HH * STATE_IN;
        const float* B1 = st_b1 + (size_t)n * HH;
        int mbase = wid * 16;
        v8f acc = gemm_a_rowmajor<STATE_IN>(W1, Xs, mbase, lane);
        for (int r = 0; r < 8; ++r) {
            int row = mbase + r + 8 * hi;
            Ys[row * 16 + col] = tanhf(acc[r] + B1[row]);
        }
    }
    __syncthreads();

    // ---- state GEMM2: h_new = tanh(W2^T(64x128) @ sh + b2); waves 4-7 stage primitives
    if (wid < 4) {
        const float* W2 = st_w2 + (size_t)n * HH * DD;
        const float* B2 = st_b2 + (size_t)n * DD;
        int mbase = wid * 16;
        v8f acc = gemm_a_transposed<DD>(W2, Ys, mbase, lane);
        for (int r = 0; r < 8; ++r) {
            int row = mbase + r + 8 * hi;
            float v = tanhf(acc[r] + B2[row]);
            if (col < BSZB) {
                out[(size_t)col * NN * DD + (size_t)n * DD + row] = v;  // h_new
                Xs[row * 16 + col] = v;                                  // mx rows 0..63
            }
        }
    } else {
        // mx rows 64..127 = new_primitives = mod_out[:, 33:97]
        for (int idx = tid - 128; idx < DD * BSZB; idx += 128) {
            int d = idx & 63, b = idx >> 6;
            Xs[(64 + d) * 16 + b] = Ms[(33 + d) * 16 + b];
        }
    }
    __syncthreads();

    // ---- msg GEMM1: mh = tanh(W1(128x128) @ mx + b1)
    {
        const float* W1 = ms_w1 + (size_t)n * HH * MSG_IN;
        const float* B1 = ms_b1 + (size_t)n * HH;
        int mbase = wid * 16;
        v8f acc = gemm_a_rowmajor<MSG_IN>(W1, Xs, mbase, lane);
        for (int r = 0; r < 8; ++r) {
            int row = mbase + r + 8 * hi;
            Ys[row * 16 + col] = tanhf(acc[r] + B1[row]);
        }
    }
    __syncthreads();

    // ---- msg GEMM2: msg = tanh(W2^T(64x128) @ mh + b2)
    if (wid < 4) {
        const float* W2 = ms_w2 + (size_t)n * HH * DD;
        const float* B2 = ms_b2 + (size_t)n * DD;
        int mbase = wid * 16;
        v8f acc = gemm_a_transposed<DD>(W2, Ys, mbase, lane);
        for (int r = 0; r < 8; ++r) {
            int row = mbase + r + 8 * hi;
            float v = tanhf(acc[r] + B2[row]);
            if (col < BSZB)
                out[(size_t)BSZB * NN * DD + (size_t)col * NN * DD + (size_t)n * DD + row] = v;
        }
    }
}

extern "C" void kernel_launch(void* const* d_in, const int* in_sizes, int n_in,
                              void* d_out, int out_size, void* d_ws, size_t ws_size,
                              hipStream_t stream) {
    (void)in_sizes; (void)n_in; (void)out_size; (void)d_ws; (void)ws_size;
    const float* h    = (const float*)d_in[0];
    const float* pm   = (const float*)d_in[1];
    const float* hebb = (const float*)d_in[2];
    const float* dec  = (const float*)d_in[3];
    const float* prim = (const float*)d_in[4];
    const float* mw1  = (const float*)d_in[5];
    const float* mb1  = (const float*)d_in[6];
    const float* mw2  = (const float*)d_in[7];
    const float* mb2  = (const float*)d_in[8];
    const float* sw1  = (const float*)d_in[9];
    const float* sb1  = (const float*)d_in[10];
    const float* sw2  = (const float*)d_in[11];
    const float* sb2  = (const float*)d_in[12];
    const float* xw1  = (const float*)d_in[13];
    const float* xb1  = (const float*)d_in[14];
    const float* xw2  = (const float*)d_in[15];
    const float* xb2  = (const float*)d_in[16];
    const float* dbw  = (const float*)d_in[17];
    const float* dgw  = (const float*)d_in[18];
    const int*   conn = (const int*)d_in[19];
    float* out = (float*)d_out;

    memgraph_kernel<<<dim3(NN), dim3(256), 0, stream>>>(
        h, pm, hebb, dec, prim,
        mw1, mb1, mw2, mb2,
        sw1, sb1, sw2, sb2,
        xw1, xb1, xw2, xb2,
        dbw, dgw, conn, out);
}